// AttentionBlock_2851858284595
// MI455X (gfx1250) — compile-verified
//
#include <hip/hip_runtime.h>
#include <hip/hip_bf16.h>

typedef float v2f __attribute__((ext_vector_type(2)));
typedef float v8f __attribute__((ext_vector_type(8)));

constexpr int   B   = 4;
constexpr int   C   = 512;
constexpr int   N   = 4096;        // 64*64
constexpr int   G   = 32;
constexpr int   CPG = C / G;       // 16
constexpr float EPS = 1e-5f;
#define RSQRT_C 0.04419417382415922f   // 1/sqrt(512)

// ---- fp32 WMMA: D(16x16) = A(16x4) * B(4x16) + C --------------------------
__device__ __forceinline__ v8f wmma4(v2f a, v2f b, v8f c) {
  return __builtin_amdgcn_wmma_f32_16x16x4_f32(false, a, false, b, (short)0, c,
                                               false, false);
}

__device__ __forceinline__ float wave_sum(float v) {
  v += __shfl_xor(v, 16);
  v += __shfl_xor(v, 8);
  v += __shfl_xor(v, 4);
  v += __shfl_xor(v, 2);
  v += __shfl_xor(v, 1);
  return v;
}

// ---------------------------------------------------------------------------
// Kernel 1: per (batch,group) mean / rstd.  Group data is contiguous: 16*N f32
// ---------------------------------------------------------------------------
__global__ __launch_bounds__(256) void gn_stats_kernel(
    const float* __restrict__ x, float* __restrict__ stats) {
  const int bg  = blockIdx.x;              // b*G + g
  const int CNT = CPG * N;                 // 65536
  const float* base = x + (size_t)bg * CNT;

  float s = 0.f, ss = 0.f;
  for (int i = threadIdx.x * 4; i < CNT; i += 256 * 4) {
    float4 v = *(const float4*)(base + i);
    s  += v.x + v.y + v.z + v.w;
    ss += v.x * v.x + v.y * v.y + v.z * v.z + v.w * v.w;
  }
  s  = wave_sum(s);
  ss = wave_sum(ss);

  __shared__ float rs[8], rss[8];
  if ((threadIdx.x & 31) == 0) { rs[threadIdx.x >> 5] = s; rss[threadIdx.x >> 5] = ss; }
  __syncthreads();
  if (threadIdx.x == 0) {
    float S = 0.f, SS = 0.f;
    for (int w = 0; w < 8; ++w) { S += rs[w]; SS += rss[w]; }
    const float mean = S / (float)CNT;
    const float var  = SS / (float)CNT - mean * mean;
    stats[2 * bg]     = mean;
    stats[2 * bg + 1] = rsqrtf(var + EPS);
  }
}

// ---------------------------------------------------------------------------
// Kernel 2: QKV GEMM with GroupNorm fused into the B-fragment loads.
//   qkv[b,o,p] = sum_c W[o,c] * norm(x[b,c,p]) + bias[o];  q rows scaled.
// Wave tile: 16(o) x 64(p); block = 8 waves -> 16 x 512.
// ---------------------------------------------------------------------------
__global__ __launch_bounds__(256) void qkv_gemm_kernel(
    const float* __restrict__ x,     const float* __restrict__ gn_w,
    const float* __restrict__ gn_b,  const float* __restrict__ stats,
    const float* __restrict__ qkv_w, const float* __restrict__ qkv_b,
    float* __restrict__ qkv) {
  const int b    = blockIdx.z;
  const int o0   = blockIdx.y * 16;
  const int wv   = threadIdx.x >> 5;
  const int lane = threadIdx.x & 31;
  const int half = lane >> 4;
  const int mr   = lane & 15;
  const int p0   = blockIdx.x * 512 + wv * 64;

  const float* xb  = x + (size_t)b * C * N;
  const float* st  = stats + (size_t)b * G * 2;
  const float* wr  = qkv_w + (size_t)(o0 + mr) * C;

  v8f a0 = {}, a1 = {}, a2 = {}, a3 = {};
  for (int k = 0; k < C; k += 4) {
    const int c0 = k + 2 * half;
    v2f af;
    af.x = wr[c0];
    af.y = wr[c0 + 1];

    const int   g    = c0 >> 4;                 // both channels in same group
    const float mean = st[2 * g], rstd = st[2 * g + 1];
    const float w0   = gn_w[c0] * rstd;
    const float bb0  = gn_b[c0] - mean * w0;
    const float w1   = gn_w[c0 + 1] * rstd;
    const float bb1  = gn_b[c0 + 1] - mean * w1;

    const float* r0 = xb + (size_t)c0 * N + p0 + mr;
    const float* r1 = r0 + N;
    v2f bf;
#define QKV_STEP(ACC, T)                                     \
    bf.x = r0[(T) * 16] * w0 + bb0;                          \
    bf.y = r1[(T) * 16] * w1 + bb1;                          \
    ACC  = wmma4(af, bf, ACC);
    QKV_STEP(a0, 0) QKV_STEP(a1, 1) QKV_STEP(a2, 2) QKV_STEP(a3, 3)
#undef QKV_STEP
  }

  float* outb = qkv + ((size_t)b * 3 * C + o0) * N;
#pragma unroll
  for (int r = 0; r < 8; ++r) {
    const int   o     = r + 8 * half;
    const int   og    = o0 + o;
    const float bias  = qkv_b[og];
    const float scale = (og < C) ? RSQRT_C : 1.0f;   // fold 1/sqrt(c) into q
    float* dst = outb + (size_t)o * N + p0 + mr;
    dst[0]  = (a0[r] + bias) * scale;
    dst[16] = (a1[r] + bias) * scale;
    dst[32] = (a2[r] + bias) * scale;
    dst[48] = (a3[r] + bias) * scale;
  }
}

// ---------------------------------------------------------------------------
// Kernel 3: softmax denominators  rsum[b,n] = sum_m exp(q[:,n]·k[:,m])
// Block owns 16 query rows; wave w sweeps keys m = w*16 + j*128.
// ---------------------------------------------------------------------------
__global__ __launch_bounds__(256) void attn_rsum_kernel(
    const float* __restrict__ qkv, float* __restrict__ rsum) {
  __shared__ float red[8 * 16];
  const int b    = blockIdx.y;
  const int n0   = blockIdx.x * 16;
  const int wv   = threadIdx.x >> 5;
  const int lane = threadIdx.x & 31;
  const int half = lane >> 4;
  const int mr   = lane & 15;

  const float* q  = qkv + (size_t)b * 3 * C * N;
  const float* kf = q + (size_t)C * N;

  float lsum[8];
#pragma unroll
  for (int r = 0; r < 8; ++r) lsum[r] = 0.f;

  for (int m0 = wv * 16; m0 < N; m0 += 128) {
    v8f s = {};
    for (int k = 0; k < C; k += 4) {
      const int c0 = k + 2 * half;
      v2f af, bf;
      af.x = q[(size_t)c0 * N + n0 + mr];          // A[n, c]
      af.y = q[(size_t)(c0 + 1) * N + n0 + mr];
      bf.x = kf[(size_t)c0 * N + m0 + mr];         // B[c, m]
      bf.y = kf[(size_t)(c0 + 1) * N + m0 + mr];
      s = wmma4(af, bf, s);
    }
#pragma unroll
    for (int r = 0; r < 8; ++r) lsum[r] += __expf(s[r]);
  }
  // reduce across the 16 lanes that hold one S row (xor <=8 stays in-half)
#pragma unroll
  for (int r = 0; r < 8; ++r) {
    float v = lsum[r];
    v += __shfl_xor(v, 1);
    v += __shfl_xor(v, 2);
    v += __shfl_xor(v, 4);
    v += __shfl_xor(v, 8);
    lsum[r] = v;
  }
  if (mr == 0) {
#pragma unroll
    for (int r = 0; r < 8; ++r) red[wv * 16 + r + 8 * half] = lsum[r];
  }
  __syncthreads();
  if (threadIdx.x < 16) {
    float s = 0.f;
    for (int w = 0; w < 8; ++w) s += red[w * 16 + threadIdx.x];
    rsum[(size_t)b * N + n0 + threadIdx.x] = s;
  }
}

// ---------------------------------------------------------------------------
// Kernel 4: out[b,c,n] = sum_m P[n,m] * v[c,m],  P = exp(S)/rsum[n]
// Block owns 16 queries; per 128-key block each wave builds one 16x16 P tile
// (WMMA S-tile -> exp -> LDS), then all 8 waves run the V*P GEMM over their
// 64 channels (4 fp32 accumulators, K=128 per block).
// ---------------------------------------------------------------------------
__global__ __launch_bounds__(256) void attn_av_kernel(
    const float* __restrict__ qkv, const float* __restrict__ rsum,
    float* __restrict__ out) {
  __shared__ float P[128 * 16];          // [m_local][n_local]
  __shared__ float rinv_s[16];
  const int b    = blockIdx.y;
  const int n0   = blockIdx.x * 16;
  const int wv   = threadIdx.x >> 5;
  const int lane = threadIdx.x & 31;
  const int half = lane >> 4;
  const int mr   = lane & 15;

  const float* q  = qkv + (size_t)b * 3 * C * N;
  const float* kf = q + (size_t)C * N;
  const float* vf = q + (size_t)2 * C * N;

  if (threadIdx.x < 16)
    rinv_s[threadIdx.x] = 1.0f / rsum[(size_t)b * N + n0 + threadIdx.x];
  __syncthreads();
  float rinv8[8];
#pragma unroll
  for (int r = 0; r < 8; ++r) rinv8[r] = rinv_s[r + 8 * half];

  v8f acc[4];
#pragma unroll
  for (int t = 0; t < 4; ++t) acc[t] = {};

  const int c0w = wv * 64;
  for (int m0 = 0; m0 < N; m0 += 128) {
    // ---- phase 1: S tile for keys [m0 + wv*16, +16) ----
    const int mcol = m0 + wv * 16;
    v8f s = {};
    for (int k = 0; k < C; k += 4) {
      const int c0 = k + 2 * half;
      v2f af, bf;
      af.x = q[(size_t)c0 * N + n0 + mr];
      af.y = q[(size_t)(c0 + 1) * N + n0 + mr];
      bf.x = kf[(size_t)c0 * N + mcol + mr];
      bf.y = kf[(size_t)(c0 + 1) * N + mcol + mr];
      s = wmma4(af, bf, s);
    }
    __syncthreads();                       // prior P consumers are done
#pragma unroll
    for (int r = 0; r < 8; ++r)
      P[(wv * 16 + mr) * 16 + r + 8 * half] = __expf(s[r]) * rinv8[r];
    __syncthreads();                       // P visible to all waves

    // ---- phase 2: acc[c,n] += V[c, m-block] * P[m-block, n] ----
#pragma unroll
    for (int t = 0; t < 4; ++t) {
      const float* vrow = vf + (size_t)(c0w + t * 16 + mr) * N + m0;
#pragma unroll 4
      for (int kk = 0; kk < 128; kk += 4) {
        const int k2 = kk + 2 * half;
        v2f af, bf;
        af.x = vrow[k2];
        af.y = vrow[k2 + 1];
        bf.x = P[k2 * 16 + mr];
        bf.y = P[(k2 + 1) * 16 + mr];
        acc[t] = wmma4(af, bf, acc[t]);
      }
    }
  }

  float* ob = out + (size_t)b * C * N;
#pragma unroll
  for (int t = 0; t < 4; ++t)
#pragma unroll
    for (int r = 0; r < 8; ++r)
      ob[(size_t)(c0w + t * 16 + r + 8 * half) * N + n0 + mr] = acc[t][r];
}

// ---------------------------------------------------------------------------
// Kernel 5: projection GEMM + bias + residual -> d_out
// ---------------------------------------------------------------------------
__global__ __launch_bounds__(256) void proj_gemm_kernel(
    const float* __restrict__ ain,   // [B, C, N] attention output
    const float* __restrict__ pw,    // [C, C]
    const float* __restrict__ pb,    // [C]
    const float* __restrict__ x,     // residual [B, C, N]
    float* __restrict__ y) {
  const int b    = blockIdx.z;
  const int o0   = blockIdx.y * 16;
  const int wv   = threadIdx.x >> 5;
  const int lane = threadIdx.x & 31;
  const int half = lane >> 4;
  const int mr   = lane & 15;
  const int p0   = blockIdx.x * 512 + wv * 64;

  const float* ab = ain + (size_t)b * C * N;
  const float* wr = pw + (size_t)(o0 + mr) * C;

  v8f a0 = {}, a1 = {}, a2 = {}, a3 = {};
  for (int k = 0; k < C; k += 4) {
    const int c0 = k + 2 * half;
    v2f af;
    af.x = wr[c0];
    af.y = wr[c0 + 1];
    const float* r0 = ab + (size_t)c0 * N + p0 + mr;
    const float* r1 = r0 + N;
    v2f bf;
#define PRJ_STEP(ACC, T)                                     \
    bf.x = r0[(T) * 16];                                     \
    bf.y = r1[(T) * 16];                                     \
    ACC  = wmma4(af, bf, ACC);
    PRJ_STEP(a0, 0) PRJ_STEP(a1, 1) PRJ_STEP(a2, 2) PRJ_STEP(a3, 3)
#undef PRJ_STEP
  }

  const float* xb = x + (size_t)b * C * N;
  float* yb = y + (size_t)b * C * N;
#pragma unroll
  for (int r = 0; r < 8; ++r) {
    const int   o    = r + 8 * half;
    const float bias = pb[o0 + o];
    const size_t row = (size_t)(o0 + o) * N + p0 + mr;
    yb[row + 0]  = xb[row + 0]  + a0[r] + bias;
    yb[row + 16] = xb[row + 16] + a1[r] + bias;
    yb[row + 32] = xb[row + 32] + a2[r] + bias;
    yb[row + 48] = xb[row + 48] + a3[r] + bias;
  }
}

// ---------------------------------------------------------------------------
extern "C" void kernel_launch(void* const* d_in, const int* in_sizes, int n_in,
                              void* d_out, int out_size, void* d_ws,
                              size_t ws_size, hipStream_t stream) {
  const float* x      = (const float*)d_in[0];
  const float* gn_w   = (const float*)d_in[1];
  const float* gn_b   = (const float*)d_in[2];
  const float* qkv_w  = (const float*)d_in[3];
  const float* qkv_b  = (const float*)d_in[4];
  const float* proj_w = (const float*)d_in[5];
  const float* proj_b = (const float*)d_in[6];
  float* out = (float*)d_out;

  char* ws = (char*)d_ws;
  const size_t SZ = (size_t)B * C * N * sizeof(float);   // 33.5 MB
  float* qkv   = (float*)(ws);                 // [B, 3C, N]      : 3*SZ
  float* attno = (float*)(ws + 3 * SZ);        // [B, C, N]       : SZ
  float* stats = (float*)(ws + 4 * SZ);        // [B*G][mean,rstd]: 1 KB
  float* rsum  = (float*)(ws + 4 * SZ + 4096); // [B, N]          : 64 KB

  gn_stats_kernel<<<B * G, 256, 0, stream>>>(x, stats);
  qkv_gemm_kernel<<<dim3(N / 512, 3 * C / 16, B), 256, 0, stream>>>(
      x, gn_w, gn_b, stats, qkv_w, qkv_b, qkv);
  attn_rsum_kernel<<<dim3(N / 16, B), 256, 0, stream>>>(qkv, rsum);
  attn_av_kernel<<<dim3(N / 16, B), 256, 0, stream>>>(qkv, rsum, attno);
  proj_gemm_kernel<<<dim3(N / 512, C / 16, B), 256, 0, stream>>>(
      attno, proj_w, proj_b, x, out);
}